// GINJKFlag_node_19524921328156
// MI455X (gfx1250) — compile-verified
//
#include <hip/hip_runtime.h>
#include <hip/hip_bf16.h>

#define GN 10000
#define GE 640000
#define GD 128
#define GH 256
#define GL 4
#define GEF 6

typedef float v2f __attribute__((ext_vector_type(2)));
typedef float v8f __attribute__((ext_vector_type(8)));

// ---------------------------------------------------------------------------
// generic zero-fill
// ---------------------------------------------------------------------------
__global__ void fill0_kernel(float* __restrict__ p, int n) {
    int i = blockIdx.x * blockDim.x + threadIdx.x;
    if (i < n) p[i] = 0.0f;
}

// ---------------------------------------------------------------------------
// degree (count of incoming edges per node) — computed once
// ---------------------------------------------------------------------------
__global__ void deg_kernel(const int* __restrict__ dst, float* __restrict__ deg, int E) {
    int i = blockIdx.x * blockDim.x + threadIdx.x;
    if (i < E) unsafeAtomicAdd(&deg[dst[i]], 1.0f);
}

__global__ void invdeg_kernel(float* __restrict__ deg, int n) {
    int i = blockIdx.x * blockDim.x + threadIdx.x;
    if (i < n) deg[i] = 1.0f / fmaxf(deg[i], 1.0f);
}

// ---------------------------------------------------------------------------
// Fused edge kernel: one wave32 per edge, 4 channels per lane.
//   ee = edge_attr[e] @ We + be   (K=6, recomputed inline: 24 FMAs/lane)
//   msg = relu(h[src] + ee)
//   ssum[dst] += msg              (global_atomic_add_f32 x4)
// Bandwidth driver: ~654 MB/layer of gather + atomic-scatter traffic.
// ---------------------------------------------------------------------------
__global__ __launch_bounds__(256) void gin_edge_kernel(
    const float* __restrict__ h, int h_stride,
    const int* __restrict__ src, const int* __restrict__ dst,
    const float* __restrict__ ea,                 // [E,6]
    const float* __restrict__ We,                 // [6,128] layer slice
    const float* __restrict__ be,                 // [128]   layer slice
    float* __restrict__ ssum, int E)
{
    int wave_id = blockIdx.x * 8 + (threadIdx.x >> 5);
    if (wave_id >= E) return;
    int lane = threadIdx.x & 31;

    int s = __builtin_amdgcn_readfirstlane(src[wave_id]);
    int d = __builtin_amdgcn_readfirstlane(dst[wave_id]);

    float a[GEF];
#pragma unroll
    for (int k = 0; k < GEF; ++k) a[k] = ea[(size_t)wave_id * GEF + k];

    int c = lane * 4;
    const float4 hv = *(const float4*)(h + (size_t)s * h_stride + c);
    float hvv[4] = {hv.x, hv.y, hv.z, hv.w};

    float* sp = ssum + (size_t)d * GD + c;
#pragma unroll
    for (int j = 0; j < 4; ++j) {
        float ee = be[c + j];
#pragma unroll
        for (int k = 0; k < GEF; ++k) ee = fmaf(a[k], We[k * GD + c + j], ee);
        float m = hvv[j] + ee;
        m = m > 0.0f ? m : 0.0f;
        unsafeAtomicAdd(sp + j, m);
    }
}

// ---------------------------------------------------------------------------
// Node MLP kernel using V_WMMA_F32_16X16X4_F32 (exact f32 math on the
// matrix pipes). One block = 16 node rows, 4 waves.
//   z = (1+eps)*h + ssum * invdeg          -> LDS  [16][132]
//   hid = relu(z @ W1 + b1)                -> LDS  [16][260]  (K=128: 32 WMMA/tile)
//   out = relu(hid @ W2 + b2)              -> out columns     (K=256: 64 WMMA/tile)
// A layout: lane&15 = M row, lane>>4 selects K pair {0,1}/{2,3}.
// B layout: lane&15 = N col, same K split. C: acc[i] -> row i + 8*(lane>>4).
// LDS row strides 132/260 (%64 == 4) -> conflict-free A reads.
// ---------------------------------------------------------------------------
__global__ __launch_bounds__(128) void gin_node_mlp_kernel(
    const float* __restrict__ h, int h_stride,
    const float* __restrict__ ssum, const float* __restrict__ invdeg,
    const float* __restrict__ W1, const float* __restrict__ b1,   // [128,256],[256]
    const float* __restrict__ W2, const float* __restrict__ b2,   // [256,128],[128]
    const float* __restrict__ eps_ptr,
    float* __restrict__ out, int out_stride, int N)
{
    __shared__ float zt[16][132];
    __shared__ float ht[16][260];

    const int t    = threadIdx.x;
    const int lane = t & 31;
    const int wave = t >> 5;
    const int row0 = blockIdx.x * 16;
    const float eps1 = 1.0f + eps_ptr[0];

    // ---- stage z tile: r = i, c = t (coalesced) ----
#pragma unroll
    for (int i = 0; i < 16; ++i) {
        int n = row0 + i;
        float v = 0.0f;
        if (n < N)
            v = eps1 * h[(size_t)n * h_stride + t] + ssum[(size_t)n * GD + t] * invdeg[n];
        zt[i][t] = v;
    }
    __syncthreads();

    const int m  = lane & 15;       // M row for A, N col for B
    const int hk = lane >> 4;       // K-pair selector

    // ---- hidden = relu(z @ W1 + b1): 16 col-tiles, 4 per wave ----
#pragma unroll
    for (int j = 0; j < 4; ++j) {
        int nb = (wave + 4 * j) * 16;
        v8f acc = {};
#pragma unroll 8
        for (int kk = 0; kk < 32; ++kk) {
            int kb = kk * 4 + hk * 2;
            v2f av, bv;
            av.x = zt[m][kb];
            av.y = zt[m][kb + 1];
            bv.x = W1[kb * GH + nb + m];
            bv.y = W1[(kb + 1) * GH + nb + m];
            acc = __builtin_amdgcn_wmma_f32_16x16x4_f32(
                false, av, false, bv, (short)0, acc, false, false);
        }
        float bias = b1[nb + m];
#pragma unroll
        for (int i = 0; i < 8; ++i) {
            float v = acc[i] + bias;
            v = v > 0.0f ? v : 0.0f;
            ht[i + 8 * hk][nb + m] = v;
        }
    }
    __syncthreads();

    // ---- out = relu(hidden @ W2 + b2): 8 col-tiles, 2 per wave ----
#pragma unroll
    for (int j = 0; j < 2; ++j) {
        int nb = (wave + 4 * j) * 16;
        v8f acc = {};
#pragma unroll 8
        for (int kk = 0; kk < 64; ++kk) {
            int kb = kk * 4 + hk * 2;
            v2f av, bv;
            av.x = ht[m][kb];
            av.y = ht[m][kb + 1];
            bv.x = W2[kb * GD + nb + m];
            bv.y = W2[(kb + 1) * GD + nb + m];
            acc = __builtin_amdgcn_wmma_f32_16x16x4_f32(
                false, av, false, bv, (short)0, acc, false, false);
        }
        float bias = b2[nb + m];
#pragma unroll
        for (int i = 0; i < 8; ++i) {
            int n = row0 + i + 8 * hk;
            float v = acc[i] + bias;
            v = v > 0.0f ? v : 0.0f;
            if (n < N) out[(size_t)n * out_stride + nb + m] = v;
        }
    }
}

// ---------------------------------------------------------------------------
// launch
// ---------------------------------------------------------------------------
extern "C" void kernel_launch(void* const* d_in, const int* in_sizes, int n_in,
                              void* d_out, int out_size, void* d_ws, size_t ws_size,
                              hipStream_t stream) {
    const float* x   = (const float*)d_in[0];
    const int*   ei  = (const int*)d_in[1];    // [2,E] row-major: src=ei, dst=ei+E
    const float* ea  = (const float*)d_in[2];  // [E,6]
    // d_in[3] = batch (unused)
    const float* We  = (const float*)d_in[4];  // [L,6,128]
    const float* be  = (const float*)d_in[5];  // [L,128]
    const float* W1  = (const float*)d_in[6];  // [L,128,256]
    const float* b1  = (const float*)d_in[7];  // [L,256]
    const float* W2  = (const float*)d_in[8];  // [L,256,128]
    const float* b2  = (const float*)d_in[9];  // [L,128]
    const float* eps = (const float*)d_in[10]; // [L]

    float* out  = (float*)d_out;               // [N, L*D]
    float* ssum = (float*)d_ws;                // N*D floats
    float* deg  = ssum + (size_t)GN * GD;      // N floats

    const int* src = ei;
    const int* dst = ei + GE;

    // degree (layer-invariant), then reciprocal in place
    fill0_kernel<<<(GN + 255) / 256, 256, 0, stream>>>(deg, GN);
    deg_kernel<<<(GE + 255) / 256, 256, 0, stream>>>(dst, deg, GE);
    invdeg_kernel<<<(GN + 255) / 256, 256, 0, stream>>>(deg, GN);

    for (int l = 0; l < GL; ++l) {
        const float* hin = (l == 0) ? x : out + (size_t)(l - 1) * GD;
        int hstride      = (l == 0) ? GD : GL * GD;

        fill0_kernel<<<(GN * GD + 255) / 256, 256, 0, stream>>>(ssum, GN * GD);

        gin_edge_kernel<<<(GE + 7) / 8, 256, 0, stream>>>(
            hin, hstride, src, dst, ea,
            We + (size_t)l * GEF * GD, be + (size_t)l * GD, ssum, GE);

        gin_node_mlp_kernel<<<(GN + 15) / 16, 128, 0, stream>>>(
            hin, hstride, ssum, deg,
            W1 + (size_t)l * GD * GH, b1 + (size_t)l * GH,
            W2 + (size_t)l * GH * GD, b2 + (size_t)l * GD,
            eps + l,
            out + (size_t)l * GD, GL * GD, GN);
    }
}